// GINConvolution_20804821581899
// MI455X (gfx1250) — compile-verified
//
#include <hip/hip_runtime.h>
#include <hip/hip_bf16.h>

// ---------------------------------------------------------------------------
// GIN convolution for MI455X (gfx1250):
//   agg[i] = sum_e val[e] * x[col[e]]  (row[e]==i)   -> atomic scatter-add
//   h  = x + agg
//   h1 = relu(h @ W1 + b1)   [N,200]
//   out = h1 @ W2 + b2       [N,64]
// MLP fused per 16-row tile using V_WMMA_F32_16X16X4_F32 (true fp32 matrix op)
// with h1 staged in LDS (never touches HBM).
// ---------------------------------------------------------------------------

#define N_NODES 50000
#define N_EDGES 800000
#define DIN     128
#define DHID    200
#define DHIDP   208          // DHID padded to multiple of 16
#define DOUT    64
#define NT1     13           // n-tiles of GEMM1 (208/16)
#define KB1     32           // k-blocks of GEMM1 (128/4)
#define KB2     52           // k-blocks of GEMM2 (208/4)

typedef __attribute__((ext_vector_type(2))) float v2f;
typedef __attribute__((ext_vector_type(8))) float v8f;

// ---------------------------------------------------------------------------
// 1) hbuf = x  (scatter-add then accumulates A@x on top -> h = x + agg)
// ---------------------------------------------------------------------------
__global__ __launch_bounds__(256)
void init_hbuf_kernel(const float* __restrict__ x, float* __restrict__ hbuf, int n4) {
    int i = blockIdx.x * blockDim.x + threadIdx.x;
    if (i < n4) {
        reinterpret_cast<float4*>(hbuf)[i] =
            reinterpret_cast<const float4*>(x)[i];
    }
}

// ---------------------------------------------------------------------------
// 2) one wave per edge; lane owns 4 contiguous features
// ---------------------------------------------------------------------------
__global__ __launch_bounds__(256)
void spmm_scatter_kernel(const float* __restrict__ x,
                         const int*   __restrict__ erow,
                         const int*   __restrict__ ecol,
                         const float* __restrict__ eval,
                         float* __restrict__ hbuf) {
    const int e = blockIdx.x * (blockDim.x >> 5) + (threadIdx.x >> 5);
    if (e >= N_EDGES) return;
    const int  lane = threadIdx.x & 31;
    const int  r = erow[e];
    const int  c = ecol[e];
    const float v = eval[e];
    const float4 xv =
        reinterpret_cast<const float4*>(x + (size_t)c * DIN)[lane];
    float* dst = hbuf + (size_t)r * DIN + lane * 4;
    atomicAdd(dst + 0, v * xv.x);
    atomicAdd(dst + 1, v * xv.y);
    atomicAdd(dst + 2, v * xv.z);
    atomicAdd(dst + 3, v * xv.w);
}

// ---------------------------------------------------------------------------
// 3) repack W1/W2 into per-lane B-fragment layout of v_wmma_f32_16x16x4_f32:
//    B frag (4x16): lanes 0-15 -> N=0..15 (K=kb*4+0/1), lanes 16-31 -> K=kb*4+2/3
//    W1p: [nt(13)][kb(32)][lane(32)] v2f, zero-padded for n >= 200
//    W2p: [nt(4)][kb(52)][lane(32)]  v2f, zero-padded for k >= 200
// ---------------------------------------------------------------------------
__global__ __launch_bounds__(256)
void repack_kernel(const float* __restrict__ W1, const float* __restrict__ b1,
                   const float* __restrict__ W2,
                   v2f* __restrict__ W1p, v2f* __restrict__ W2p,
                   float* __restrict__ b1p) {
    const int i    = blockIdx.x * blockDim.x + threadIdx.x;
    const int lane = i & 31;
    const int hsel = lane >> 4;
    const int lrow = lane & 15;
    if (i < NT1 * KB1 * 32) {
        const int nt = i / (KB1 * 32);
        const int kb = (i >> 5) % KB1;
        const int n  = nt * 16 + lrow;
        const int k0 = kb * 4 + hsel * 2;
        v2f v;
        v.x = (n < DHID) ? W1[(size_t)k0 * DHID + n] : 0.0f;
        v.y = (n < DHID) ? W1[(size_t)(k0 + 1) * DHID + n] : 0.0f;
        W1p[i] = v;
    }
    if (i < 4 * KB2 * 32) {
        const int nt = i / (KB2 * 32);
        const int kb = (i >> 5) % KB2;
        const int n  = nt * 16 + lrow;
        const int k0 = kb * 4 + hsel * 2;
        v2f v;
        v.x = (k0 < DHID)     ? W2[(size_t)k0 * DOUT + n] : 0.0f;
        v.y = (k0 + 1 < DHID) ? W2[(size_t)(k0 + 1) * DOUT + n] : 0.0f;
        W2p[i] = v;
    }
    if (i < DHIDP) b1p[i] = (i < DHID) ? b1[i] : 0.0f;
}

// ---------------------------------------------------------------------------
// 4) fused MLP: 16 rows per workgroup, 4 waves, fp32 WMMA; h1 lives in LDS
// ---------------------------------------------------------------------------
__global__ __launch_bounds__(128)
void gin_mlp_kernel(const float* __restrict__ hbuf,
                    const v2f*  __restrict__ W1p,
                    const v2f*  __restrict__ W2p,
                    const float* __restrict__ b1p,
                    const float* __restrict__ b2,
                    float* __restrict__ out) {
    __shared__ float hS[16 * DHIDP];

    const int m0   = blockIdx.x * 16;
    const int lane = threadIdx.x & 31;
    const int wave = threadIdx.x >> 5;
    const int hsel = lane >> 4;     // selects K pair (0: k+0/1, 1: k+2/3)
    const int lrow = lane & 15;     // A: M row / B,C,D: N column

    // --- preload A fragments of the 16x128 (x+agg) tile into registers ----
    const float* Arow = hbuf + (size_t)(m0 + lrow) * DIN;
    v2f aFr[KB1];
#pragma unroll
    for (int kb = 0; kb < KB1; ++kb) {
        const int k0 = kb * 4 + hsel * 2;
        v2f a;
        a.x = Arow[k0];
        a.y = Arow[k0 + 1];
        aFr[kb] = a;
    }

    // --- GEMM1 + bias + ReLU -> LDS (16 x 208) ----------------------------
    for (int nt = wave; nt < NT1; nt += 4) {
        const float bias = b1p[nt * 16 + lrow];
        v8f c;
#pragma unroll
        for (int j = 0; j < 8; ++j) c[j] = bias;
        const v2f* Bp = W1p + (size_t)nt * KB1 * 32 + lane;
#pragma unroll
        for (int kb = 0; kb < KB1; ++kb) {
            v2f b = Bp[kb * 32];
            c = __builtin_amdgcn_wmma_f32_16x16x4_f32(
                    false, aFr[kb], false, b, (short)0, c, false, false);
        }
#pragma unroll
        for (int j = 0; j < 8; ++j) {
            // D layout: lanes 0-15 -> M=j, lanes 16-31 -> M=j+8; N = nt*16+lrow
            hS[(j + 8 * hsel) * DHIDP + nt * 16 + lrow] = fmaxf(c[j], 0.0f);
        }
    }
    __syncthreads();

    // --- GEMM2: each wave one 16-col tile of out (4 tiles = 64 cols) ------
    {
        const int nt = wave;
        const float bias = b2[nt * 16 + lrow];
        v8f c;
#pragma unroll
        for (int j = 0; j < 8; ++j) c[j] = bias;
        const v2f* Bp = W2p + (size_t)nt * KB2 * 32 + lane;
#pragma unroll
        for (int kb = 0; kb < KB2; ++kb) {
            const int k0 = kb * 4 + hsel * 2;
            v2f a;
            a.x = hS[lrow * DHIDP + k0];
            a.y = hS[lrow * DHIDP + k0 + 1];
            v2f b = Bp[kb * 32];
            c = __builtin_amdgcn_wmma_f32_16x16x4_f32(
                    false, a, false, b, (short)0, c, false, false);
        }
#pragma unroll
        for (int j = 0; j < 8; ++j) {
            out[(size_t)(m0 + j + 8 * hsel) * DOUT + nt * 16 + lrow] = c[j];
        }
    }
}

// ---------------------------------------------------------------------------
// launch
// ---------------------------------------------------------------------------
extern "C" void kernel_launch(void* const* d_in, const int* in_sizes, int n_in,
                              void* d_out, int out_size, void* d_ws, size_t ws_size,
                              hipStream_t stream) {
    const float* x    = (const float*)d_in[0];
    const int*   erow = (const int*)d_in[1];
    const int*   ecol = (const int*)d_in[2];
    const float* eval = (const float*)d_in[3];
    const float* W1   = (const float*)d_in[4];
    const float* b1   = (const float*)d_in[5];
    const float* W2   = (const float*)d_in[6];
    const float* b2   = (const float*)d_in[7];
    float* out = (float*)d_out;

    char* ws = (char*)d_ws;
    float* hbuf = (float*)(ws + 0);                       // 50000*128*4 = 25,600,000 B
    v2f*   W1p  = (v2f*)  (ws + 25600000);                // 13*32*32*8  =    106,496 B
    v2f*   W2p  = (v2f*)  (ws + 25600000 + 106496);       //  4*52*32*8  =     53,248 B
    float* b1p  = (float*)(ws + 25600000 + 106496 + 53248); // 208*4

    // 1) hbuf = x
    const int n4 = N_NODES * DIN / 4;                     // 1,600,000 float4
    init_hbuf_kernel<<<(n4 + 255) / 256, 256, 0, stream>>>(x, hbuf, n4);

    // 2) repack weights into WMMA B-fragment layout (tiny, overlap-free on stream)
    repack_kernel<<<(NT1 * KB1 * 32 + 255) / 256, 256, 0, stream>>>(
        W1, b1, W2, W1p, W2p, b1p);

    // 3) scatter-add: hbuf += A @ x   (one wave per edge)
    spmm_scatter_kernel<<<(N_EDGES * 32) / 256, 256, 0, stream>>>(
        x, erow, ecol, eval, hbuf);

    // 4) fused MLP with fp32 WMMA
    gin_mlp_kernel<<<N_NODES / 16, 128, 0, stream>>>(hbuf, W1p, W2p, b1p, b2, out);
}